// TemporalCoding_1297080123575
// MI455X (gfx1250) — compile-verified
//
#include <hip/hip_runtime.h>
#include <cstdint>
#include <cstddef>

// Temporal coding: x (32, 32768) f32 -> spikes (32, 32, 32768) f32 one-hot in time.
// Pure store-bandwidth problem (4 MB in, 128 MB out); no matrix math -> no WMMA.
// Uses CDNA5 async global->LDS copy (ASYNCcnt) for the input staging path.

#define TC_TIMESTEPS 32
#define TC_F 32768
#define TC_FV (TC_F / 4)          // 8192 float4 per (b,t) row
#define TC_BLOCK 256

__global__ __launch_bounds__(TC_BLOCK)
void temporal_coding_kernel(const float* __restrict__ x, float4* __restrict__ out) {
    __shared__ float4 tile[TC_BLOCK];

    const unsigned tid = threadIdx.x;
    const unsigned gv  = blockIdx.x * TC_BLOCK + tid;   // float4 index into x

    // ---- CDNA5 async copy: global -> LDS (bypasses VGPRs, tracked by ASYNCcnt)
    unsigned lds_addr = (unsigned)(uintptr_t)&tile[tid];          // low 32 bits = LDS offset
    unsigned voff     = gv * 16u;                                  // byte offset (input is 4 MB, fits u32)
    unsigned long long sbase = (unsigned long long)(uintptr_t)x;   // SGPR pair base
    asm volatile("global_load_async_to_lds_b128 %0, %1, %2"
                 :: "v"(lds_addr), "v"(voff), "s"(sbase)
                 : "memory");
    asm volatile("s_wait_asynccnt 0" ::: "memory");
    // Each lane reads back only the bytes it fetched -> no barrier needed.
    float4 xv = tile[tid];

    // ---- spike time per element, packed as a 32-bit one-hot mask
    const float avail = (float)(TC_TIMESTEPS - 1);  // MIN_DELAY = 0
    unsigned oh[4];
    {
        float xs[4] = {xv.x, xv.y, xv.z, xv.w};
        #pragma unroll
        for (int i = 0; i < 4; ++i) {
            float xn = fminf(fmaxf(xs[i], 0.0f), 1.0f);
            int st = (int)((1.0f - xn) * avail);     // truncation, matches .astype(int32)
            st = st < 0 ? 0 : (st > TC_TIMESTEPS - 1 ? TC_TIMESTEPS - 1 : st);
            oh[i] = (xn > 0.0f) ? (1u << st) : 0u;
        }
    }

    // ---- 32 coalesced b128 stores (one per timestep) — the bandwidth-bound part
    const unsigned b  = gv / TC_FV;                 // batch index
    const unsigned fv = gv % TC_FV;                 // float4 column
    float4* dst = out + (size_t)b * (size_t)(TC_TIMESTEPS * TC_FV) + fv;

    #pragma unroll
    for (int t = 0; t < TC_TIMESTEPS; ++t) {
        float4 v;
        v.x = (float)((oh[0] >> t) & 1u);
        v.y = (float)((oh[1] >> t) & 1u);
        v.z = (float)((oh[2] >> t) & 1u);
        v.w = (float)((oh[3] >> t) & 1u);
        dst[(size_t)t * TC_FV] = v;
    }
}

extern "C" void kernel_launch(void* const* d_in, const int* in_sizes, int n_in,
                              void* d_out, int out_size, void* d_ws, size_t ws_size,
                              hipStream_t stream) {
    (void)n_in; (void)out_size; (void)d_ws; (void)ws_size;
    const float* x = (const float*)d_in[0];
    float4* out    = (float4*)d_out;

    const int n  = in_sizes[0];       // 32 * 32768 = 1048576 elements
    const int n4 = n / 4;             // 262144 float4s
    dim3 grid(n4 / TC_BLOCK);         // 1024 blocks of 256 threads (8 wave32 waves)
    temporal_coding_kernel<<<grid, TC_BLOCK, 0, stream>>>(x, out);
}